// BPMLLLoss_83648783057226
// MI455X (gfx1250) — compile-verified
//
#include <hip/hip_runtime.h>

typedef __attribute__((ext_vector_type(2))) float v2f;
typedef __attribute__((ext_vector_type(8))) float v8f;

#define BB 32
#define LL 2048
#define KCHUNKS 4   // K-chunks (waves per row-tile)
#define TILES 2     // row tiles of 16 rows

__global__ __launch_bounds__(256)
void bpmll_wmma_kernel(const float* __restrict__ c,
                       const int*   __restrict__ y,
                       float* __restrict__ out)
{
    __shared__ float sPos[TILES][KCHUNKS][16];
    __shared__ float sNeg[TILES][KCHUNKS][16];
    __shared__ float sCnt[TILES][KCHUNKS][16];
    __shared__ float sRow[BB];

    const int tid  = threadIdx.x;
    const int wave = tid >> 5;      // 0..7 (wave32)
    const int lane = tid & 31;
    const int tile = wave & 1;      // which 16-row tile
    const int kc   = wave >> 1;     // which K chunk (0..3), 512 cols each

    // A-matrix 16x4 f32 layout: lanes 0-15 hold K={0,1}, lanes 16-31 K={2,3}
    const int half = lane >> 4;
    const int m    = lane & 15;     // row within tile
    const int row  = tile * 16 + m;

    const float* crow = c + row * LL;
    const int*   yrow = y + row * LL;

    v8f accPos = {};   // sum_{k in Y}    exp(-c_k)   per row
    v8f accNeg = {};   // sum_{l in Ybar} exp(+c_l)   per row
    v8f accCnt = {};   // |Y| per row
    v2f bOnes;  bOnes.x = 1.0f;  bOnes.y = 1.0f;   // ones B-matrix (layout-free)

    const int k0 = kc * (LL / KCHUNKS);
    const int k1 = k0 + (LL / KCHUNKS);
    for (int k = k0; k < k1; k += 4) {
        // 128-bit aligned loads; this lane consumes 2 of the 4 elements
        const float4 cv = *reinterpret_cast<const float4*>(crow + k);
        const int4   yv = *reinterpret_cast<const int4*>(yrow + k);
        const float c0 = half ? cv.z : cv.x;
        const float c1 = half ? cv.w : cv.y;
        const int   y0 = half ? yv.z : yv.x;
        const int   y1 = half ? yv.w : yv.y;

        v2f aPos, aNeg, aCnt;
        aPos.x = (y0 == 1) ? __expf(-c0) : 0.0f;
        aPos.y = (y1 == 1) ? __expf(-c1) : 0.0f;
        aNeg.x = (y0 != 1) ? __expf(c0)  : 0.0f;
        aNeg.y = (y1 != 1) ? __expf(c1)  : 0.0f;
        aCnt.x = (y0 == 1) ? 1.0f : 0.0f;
        aCnt.y = (y1 == 1) ? 1.0f : 0.0f;

        // D = A * ones + C  -> per-row running sums, exact f32 accumulation
        accPos = __builtin_amdgcn_wmma_f32_16x16x4_f32(
            false, aPos, false, bOnes, (short)0, accPos, false, false);
        accNeg = __builtin_amdgcn_wmma_f32_16x16x4_f32(
            false, aNeg, false, bOnes, (short)0, accNeg, false, false);
        accCnt = __builtin_amdgcn_wmma_f32_16x16x4_f32(
            false, aCnt, false, bOnes, (short)0, accCnt, false, false);
    }

    // C/D layout: VGPR r, lane 0 -> row r; lane 16 -> row 8+r (all N equal)
    if (m == 0) {
        const int base = half * 8;
        #pragma unroll
        for (int r = 0; r < 8; ++r) {
            sPos[tile][kc][base + r] = accPos[r];
            sNeg[tile][kc][base + r] = accNeg[r];
            sCnt[tile][kc][base + r] = accCnt[r];
        }
    }
    __syncthreads();

    // combine the 4 K-chunk partials per row, form per-row loss term
    if (tid < BB) {
        const int t  = tid >> 4;
        const int rr = tid & 15;
        float pos = 0.0f, neg = 0.0f, cnt = 0.0f;
        #pragma unroll
        for (int q = 0; q < KCHUNKS; ++q) {
            pos += sPos[t][q][rr];
            neg += sNeg[t][q][rr];
            cnt += sCnt[t][q][rr];
        }
        const float denom = cnt * ((float)LL - cnt);   // |Y|^1 * |Ybar|^1
        sRow[tid] = (pos * neg) / denom;
    }
    __syncthreads();

    if (tid == 0) {
        float s = 0.0f;
        #pragma unroll 4
        for (int i = 0; i < BB; ++i) s += sRow[i];
        out[0] = s / (float)BB;
    }
}

extern "C" void kernel_launch(void* const* d_in, const int* in_sizes, int n_in,
                              void* d_out, int out_size, void* d_ws, size_t ws_size,
                              hipStream_t stream) {
    const float* c = (const float*)d_in[0];   // [32, 2048] f32
    const int*   y = (const int*)d_in[1];     // [32, 2048] i32
    float* out = (float*)d_out;               // scalar f32
    (void)in_sizes; (void)n_in; (void)out_size; (void)d_ws; (void)ws_size;
    bpmll_wmma_kernel<<<1, 256, 0, stream>>>(c, y, out);
}